// ParallelLinear_77627238908437
// MI455X (gfx1250) — compile-verified
//
#include <hip/hip_runtime.h>
#include <hip/hip_bf16.h>

typedef __attribute__((ext_vector_type(2))) float v2f;
typedef __attribute__((ext_vector_type(8))) float v8f;

// out[e] (O x T) = W[e] (O x N) @ x[e] (N x T) + b[e]
// E=10000, N=O=64, T=256. One workgroup (8 wave32) per edge.
// Wave w: o-tile = w&3 (rows ob..ob+15), t-tiles (w>>2)*8 .. +7 (16 cols each).
// WMMA: D(16x16,f32) = A(16x4,f32) x B(4x16,f32) + C, K-loop of 16 steps.

__global__ __launch_bounds__(256) void ParallelLinear_77627238908437_kernel(
    const float* __restrict__ x,   // [E, N, T]
    const float* __restrict__ W,   // [E, O, N]
    const float* __restrict__ b,   // [E, O]
    float* __restrict__ out)       // [E, O, T]
{
    constexpr int N = 64, O = 64, T = 256;

    const int e    = blockIdx.x;
    const int tid  = threadIdx.x;      // 0..255
    const int wave = tid >> 5;         // 0..7
    const int lane = tid & 31;
    const int half = lane >> 4;        // 0 or 1 (lane half)
    const int l16  = lane & 15;

    const float* __restrict__ xe = x + (size_t)e * N * T;
    const float* __restrict__ We = W + (size_t)e * O * N;
    const float* __restrict__ be = b + (size_t)e * O;
    float* __restrict__ oe       = out + (size_t)e * O * T;

    const int ob  = (wave & 3) * 16;   // output-row tile base
    const int tb0 = (wave >> 2) * 128; // first t column for this wave

    // ---- Preload A fragments (W tile rows ob..ob+15, all K=0..63) ----
    // ISA layout, 32-bit A 16x4: lanes 0-15 hold M=0..15;
    //   V0 = K = k0 + 2*half, V1 = K = k0 + 2*half + 1.
    v2f afrag[16];
    {
        const float* wrow = We + (ob + l16) * N + 2 * half;
        #pragma unroll
        for (int ks = 0; ks < 16; ++ks) {
            afrag[ks].x = wrow[ks * 4 + 0];
            afrag[ks].y = wrow[ks * 4 + 1];
        }
    }

    // Bias values for this wave's 8 C/D VGPR rows (per-lane-half row mapping).
    float bias[8];
    #pragma unroll
    for (int v = 0; v < 8; ++v)
        bias[v] = be[ob + v + 8 * half];

    // ---- Loop over this wave's 8 t-tiles ----
    #pragma unroll 2
    for (int j = 0; j < 8; ++j) {
        const int tb = tb0 + j * 16;
        v8f acc = {};

        // B fragment base: X[k0 + 2*half][tb + l16], V1 one row below.
        const float* xcol = xe + (2 * half) * T + tb + l16;

        #pragma unroll
        for (int ks = 0; ks < 16; ++ks) {
            v2f bfrag;
            bfrag.x = xcol[ks * 4 * T];
            bfrag.y = xcol[ks * 4 * T + T];
            acc = __builtin_amdgcn_wmma_f32_16x16x4_f32(
                /*neg_a=*/false, afrag[ks],
                /*neg_b=*/false, bfrag,
                /*c_mod=*/(short)0, acc,
                /*reuse_a=*/false, /*reuse_b=*/false);
        }

        // C/D layout: lane L, VGPR v -> row (ob + v + 8*half), col tb + l16.
        #pragma unroll
        for (int v = 0; v < 8; ++v) {
            const int row = ob + v + 8 * half;
            oe[row * T + tb + l16] = acc[v] + bias[v];
        }
    }
}

extern "C" void kernel_launch(void* const* d_in, const int* in_sizes, int n_in,
                              void* d_out, int out_size, void* d_ws, size_t ws_size,
                              hipStream_t stream) {
    constexpr int N = 64, T = 256;
    const float* x = (const float*)d_in[0];
    const float* W = (const float*)d_in[1];
    const float* b = (const float*)d_in[2];
    float* out     = (float*)d_out;

    const int E = in_sizes[0] / (N * T);  // 10000

    dim3 grid(E);
    dim3 block(256);
    hipLaunchKernelGGL(ParallelLinear_77627238908437_kernel,
                       grid, block, 0, stream, x, W, b, out);
}